// MHCGNet_43104291783155
// MI455X (gfx1250) — compile-verified
//
#include <hip/hip_runtime.h>
#include <stdint.h>

// MIMO MHCGNet detector, CDNA5 (gfx1250) port.
// MR=64 rx, NT=32 tx, S=32 chains, ITER=32, BS=512, 16QAM.
#define MRX   64
#define NTX   32
#define SCH   32
#define ITERS 32
#define BSZ   512
#define DQAMF 0.31622776601683794f   // sqrt(3/(2*15))

typedef __attribute__((ext_vector_type(2))) float v2f;
typedef __attribute__((ext_vector_type(8))) float v8f;

// ---------------- deterministic stateless RNG ----------------
__device__ __forceinline__ unsigned int mix32(unsigned int x) {
  x ^= x >> 16; x *= 0x7feb352du;
  x ^= x >> 15; x *= 0x846ca68bu;
  x ^= x >> 16; return x;
}
__device__ __forceinline__ float u01(unsigned int h) {
  return (float)(h >> 8) * (1.0f / 16777216.0f) + (0.5f / 16777216.0f);
}
__device__ __forceinline__ float gauss1(unsigned int key) {
  unsigned int h1 = mix32(key);
  unsigned int h2 = mix32(key ^ 0x9E3779B9u);
  float r = sqrtf(-2.0f * __logf(u01(h1)));
  return r * __cosf(6.28318530718f * u01(h2));
}

// ---------------- WMMA f32 16x16x4 fragment helpers (wave32) ----------------
// A (16x4): lanes 0-15 hold M=0..15 K={0,1}; lanes 16-31 hold K={2,3}.
__device__ __forceinline__ v2f frag_a(const float* M, int ld, int mr, int kk, int lane) {
  int m = lane & 15, kh = (lane >> 4) << 1;
  v2f a; a.x = M[(mr + m) * ld + kk + kh]; a.y = M[(mr + m) * ld + kk + kh + 1];
  return a;
}
// A-transposed access: element [m][k] = M[k*ld + m]
__device__ __forceinline__ v2f frag_aT(const float* M, int ld, int mr, int kk, int lane) {
  int m = lane & 15, kh = (lane >> 4) << 1;
  v2f a; a.x = M[(kk + kh) * ld + mr + m]; a.y = M[(kk + kh + 1) * ld + mr + m];
  return a;
}
// B (4x16): mirror of A — lane holds col N=lane&15, K pair by lane half.
__device__ __forceinline__ v2f frag_b(const float* B, int ld, int kk, int nc, int lane) {
  int n = lane & 15, kh = (lane >> 4) << 1;
  v2f b; b.x = B[(kk + kh) * ld + nc + n]; b.y = B[(kk + kh + 1) * ld + nc + n];
  return b;
}

// Tiled GEMM D[M,N] = A[M,K] (optionally A^T) * B[K,N], 8 waves round-robin.
__device__ __forceinline__ void wmma_gemm(const float* A, int lda, bool transA,
                                          const float* B, int ldb,
                                          float* D, int ldd,
                                          int M, int N, int K,
                                          int wave, int lane) {
  int tilesN = N >> 4;
  int ntiles = (M >> 4) * tilesN;
  for (int t = wave; t < ntiles; t += 8) {
    int tm = (t / tilesN) << 4, tn = (t % tilesN) << 4;
    v8f c = {};
    for (int kk = 0; kk < K; kk += 4) {
      v2f a = transA ? frag_aT(A, lda, tm, kk, lane) : frag_a(A, lda, tm, kk, lane);
      v2f b = frag_b(B, ldb, kk, tn, lane);
      c = __builtin_amdgcn_wmma_f32_16x16x4_f32(false, a, false, b, (short)0, c,
                                                false, false);
    }
    int n = lane & 15, mh = (lane >> 4) << 3;
    for (int v = 0; v < 8; ++v)
      D[(tm + mh + v) * ldd + tn + n] = c[v];
  }
}

// rn[s] += sum_i (Y2[i] - (A2 @ X)[i][s])^2  ; A2 is 128x64, X is 64x32.
// Residual kept in registers; norms reduced with LDS float atomics.
__device__ __forceinline__ void resid_norms(const float* A2, const float* X,
                                            const float* Y2, float* rn,
                                            int wave, int lane) {
  for (int t = wave; t < 16; t += 8) {
    int tm = (t >> 1) << 4, tn = (t & 1) << 4;
    v8f c = {};
    for (int kk = 0; kk < 64; kk += 4) {
      v2f a = frag_a(A2, 64, tm, kk, lane);
      v2f b = frag_b(X, 32, kk, tn, lane);
      c = __builtin_amdgcn_wmma_f32_16x16x4_f32(false, a, false, b, (short)0, c,
                                                false, false);
    }
    int n = lane & 15, mh = (lane >> 4) << 3;
    float acc = 0.0f;
    for (int v = 0; v < 8; ++v) {
      float d = Y2[tm + mh + v] - c[v];
      acc += d * d;
    }
    atomicAdd(&rn[tn + n], acc);
  }
}

// ---------------- dynamic LDS layout (floats) ----------------
#define OF_A2   0        // 128*64 realified channel [[Ar,-Ai],[Ai,Ar]]
#define OF_XI2  8192     // 64*64  realified (A^H A + nv I)
#define OF_Y2   12288    // 128    [yr; yi]
#define OF_YT2  12416    // 64     realified A^H y
#define OF_XT2  12480    // 64     realified true x
#define OF_XHAT 12544    // 2048   (64 rows x 32 chains each)
#define OF_XSV  14592
#define OF_XCG  16640
#define OF_RCG  18688
#define OF_DI   20736
#define OF_XPP  22784
#define OF_TMP  24832    // xi@di scratch
#define OF_SC   26880    // scalar arrays
#define SMEM_FLOATS (26880 + 12 * 32)

extern "C" __global__ void __launch_bounds__(256)
mhcg_zero(float* out, int out_size) {
  int base = BSZ * NTX * 2;
  int i = base + (int)threadIdx.x;
  if ((int)threadIdx.x < 1 + ITERS && i < out_size) out[i] = 0.0f;
}

extern "C" __global__ void __launch_bounds__(256)
mhcg_kernel(const double* __restrict__ Xtrue, const double* __restrict__ Yin,
            const double* __restrict__ Ain, const double* __restrict__ NvIn,
            float* __restrict__ out, int out_size) {
  extern __shared__ float smem[];
  float* A2   = smem + OF_A2;
  float* Xi2  = smem + OF_XI2;
  float* Y2   = smem + OF_Y2;
  float* YT2  = smem + OF_YT2;
  float* XT2  = smem + OF_XT2;
  float* xhat = smem + OF_XHAT;
  float* xsv  = smem + OF_XSV;
  float* xcg  = smem + OF_XCG;
  float* rcg  = smem + OF_RCG;
  float* di   = smem + OF_DI;
  float* xpp  = smem + OF_XPP;
  float* tmpM = smem + OF_TMP;
  float* rn   = smem + OF_SC;       // accepted residual norms
  float* rnsv = rn + 32;            // survivor norms
  float* rncg = rnsv + 32;          // CG residual norm
  float* rnew = rncg + 32;
  float* rnpp = rnew + 32;          // proposal norms
  float* den  = rnpp + 32;
  float* alf  = den + 32;
  float* bet  = alf + 32;
  float* stp  = bet + 32;
  float* fsv  = stp + 32;           // survivor-update flag
  float* fac  = fsv + 32;           // metropolis-accept flag
  float* msep = fac + 32;           // [0]: block mse partial
  __shared__ int sminS;

  const int b = blockIdx.x;
  const int tid = threadIdx.x;
  const int wave = tid >> 5;
  const int lane = tid & 31;

  // ---- load & realify inputs (fp64 -> fp32) ----
  const double* Ab = Ain + (size_t)b * MRX * NTX * 2;
  for (int idx = tid; idx < MRX * NTX; idx += 256) {
    __builtin_prefetch(&Ab[(size_t)(idx + 512) * 2], 0, 0);
    int i = idx >> 5, j = idx & 31;
    float ar = (float)Ab[(size_t)idx * 2 + 0];
    float ai = (float)Ab[(size_t)idx * 2 + 1];
    A2[i * 64 + j]           = ar;
    A2[i * 64 + 32 + j]      = -ai;
    A2[(i + 64) * 64 + j]    = ai;
    A2[(i + 64) * 64 + 32 + j] = ar;
  }
  const double* Yb = Yin + (size_t)b * MRX * 2;
  for (int i = tid; i < MRX; i += 256) {
    Y2[i]      = (float)Yb[i * 2 + 0];
    Y2[64 + i] = (float)Yb[i * 2 + 1];
  }
  const double* Xb = Xtrue + (size_t)b * NTX * 2;
  for (int j = tid; j < NTX; j += 256) {
    XT2[j]      = (float)Xb[j * 2 + 0];
    XT2[32 + j] = (float)Xb[j * 2 + 1];
  }
  if (tid < 32) { rn[tid] = 0.0f; rncg[tid] = 0.0f; }
  __syncthreads();

  // ---- Xi2 = A2^T A2 (+ nv I): 64x128x64 WMMA GEMM ----
  wmma_gemm(A2, 64, true, A2, 64, Xi2, 64, 64, 64, 128, wave, lane);
  __syncthreads();
  float nv = (float)NvIn[0];
  if (tid < 64) {
    Xi2[tid * 64 + tid] += nv;
    float sacc = 0.0f;                       // YT2 = A2^T [yr; yi]
    for (int i = 0; i < 128; ++i) sacc += A2[i * 64 + tid] * Y2[i];
    YT2[tid] = sacc;
  }

  // ---- random constellation init ----
  for (int idx = tid; idx < NTX * SCH; idx += 256) {
    int j = idx >> 5, s = idx & 31;
    unsigned int h = mix32(((unsigned)b * 2246822519u) ^
                           ((unsigned)idx * 3266489917u) ^ 0x36E5A1CEu);
    float re = (2.0f * (float)((h >> 2) & 3u) - 3.0f) * DQAMF;
    float im = (2.0f * (float)(h & 3u) - 3.0f) * DQAMF;
    xhat[j * 32 + s] = re; xhat[(j + 32) * 32 + s] = im;
    xcg [j * 32 + s] = re; xcg [(j + 32) * 32 + s] = im;
    xsv [j * 32 + s] = re; xsv [(j + 32) * 32 + s] = im;
  }
  __syncthreads();

  resid_norms(A2, xhat, Y2, rn, wave, lane);      // rn = ||y - A xhat||^2
  __syncthreads();
  if (tid < 32) rnsv[tid] = rn[tid];

  // ---- CG init: r_cg = ytilde - Xi2 @ xcg ----
  wmma_gemm(Xi2, 64, false, xcg, 32, tmpM, 32, 64, 32, 64, wave, lane);
  __syncthreads();
  for (int idx = tid; idx < 64 * SCH; idx += 256) {
    int s = idx & 31, r = idx >> 5;
    float v = YT2[r] - tmpM[idx];
    rcg[idx] = v; di[idx] = v;
    atomicAdd(&rncg[s], v * v);
  }

  // ---------------- main MCMC+CG loop ----------------
  for (int t = 0; t < ITERS; ++t) {
    __syncthreads();
    if (tid < 32) { rnew[tid] = 0.0f; rnpp[tid] = 0.0f; den[tid] = 0.0f; }
    if (tid == 0) msep[0] = 0.0f;
    // xi_di = Xi2 @ di   (64x64x32 WMMA GEMM, 1 tile/wave)
    wmma_gemm(Xi2, 64, false, di, 32, tmpM, 32, 64, 32, 64, wave, lane);
    __syncthreads();
    // denom_s = <di, xi_di> per column  (== Re<conj(di), xi di>)
    for (int idx = tid; idx < 64 * SCH; idx += 256)
      atomicAdd(&den[idx & 31], di[idx] * tmpM[idx]);
    __syncthreads();
    if (tid < 32) {
      alf[tid] = rncg[tid] / den[tid];
      stp[tid] = fmaxf(DQAMF, sqrtf(rn[tid]) * (1.0f / 32.0f));
    }
    __syncthreads();
    // CG step + Gaussian proposal + grid quantization (per-axis nearest)
    for (int idx = tid; idx < 64 * SCH; idx += 256) {
      int s = idx & 31;
      float xc = xcg[idx] + alf[s] * di[idx];
      xcg[idx] = xc;
      unsigned int key = ((unsigned)b * 73856093u) ^ ((unsigned)t * 19349663u) ^
                         ((unsigned)idx * 83492791u) ^ 0x13579BDFu;
      float z = xc + stp[s] * gauss1(key) * 0.70710678f;
      float q = rintf((z * (1.0f / DQAMF) + 3.0f) * 0.5f);
      q = fminf(fmaxf(q, 0.0f), 3.0f);
      xpp[idx] = (2.0f * q - 3.0f) * DQAMF;
    }
    __syncthreads();
    // proposal residual norms via 128x64x32 WMMA GEMM (residual never stored)
    resid_norms(A2, xpp, Y2, rnpp, wave, lane);
    __syncthreads();
    // survivor + Metropolis decisions (per chain)
    if (tid < 32) {
      float rp = rnpp[tid];
      float svd = (rp < rnsv[tid]) ? 1.0f : 0.0f;
      fsv[tid] = svd;
      if (svd != 0.0f) rnsv[tid] = rp;
      float p = __expf(fminf(0.0f, -(rp - rn[tid]) * 0.5f));
      unsigned int ku = mix32(((unsigned)b * 2654435761u) ^
                              ((unsigned)t * 40503u) ^ (unsigned)tid ^ 0xCAFEBABEu);
      float am = (p >= u01(ku)) ? 1.0f : 0.0f;
      fac[tid] = am;
      if (am != 0.0f) rn[tid] = rp;
    }
    __syncthreads();
    // apply decisions, CG residual update, mse partial
    float mseAcc = 0.0f;
    for (int idx = tid; idx < 64 * SCH; idx += 256) {
      int s = idx & 31;
      float xp = xpp[idx];
      if (fsv[s] != 0.0f) xsv[idx] = xp;
      float xh = xhat[idx];
      if (fac[s] != 0.0f) { xh = xp; xhat[idx] = xh; }
      float d = xh - XT2[idx >> 5];
      mseAcc += d * d;
      float rc = rcg[idx] - alf[s] * tmpM[idx];
      rcg[idx] = rc;
      atomicAdd(&rnew[s], rc * rc);
    }
    atomicAdd(&msep[0], mseAcc);
    __syncthreads();
    if (tid < 32) { bet[tid] = rnew[tid] / rncg[tid]; rncg[tid] = rnew[tid]; }
    if (tid == 0) {
      int mo = BSZ * NTX * 2 + 1 + t;
      if (mo < out_size)
        atomicAdd(&out[mo], msep[0] * (1.0f / (float)(BSZ * SCH * NTX)));
    }
    __syncthreads();
    for (int idx = tid; idx < 64 * SCH; idx += 256)
      di[idx] = rcg[idx] + bet[idx & 31] * di[idx];
  }

  // ---------------- epilogue: best chain, loss, xhat_out ----------------
  __syncthreads();
  if (tid == 0) {
    float rmin = rnsv[0]; int sm = 0;
    for (int s = 1; s < 32; ++s)
      if (rnsv[s] < rmin) { rmin = rnsv[s]; sm = s; }
    sminS = sm;
    int li = BSZ * NTX * 2;
    if (li < out_size) atomicAdd(&out[li], sqrtf(rmin) * (1.0f / (float)BSZ));
  }
  __syncthreads();
  if (tid < NTX) {
    int sm = sminS;
    int oi = (b * NTX + tid) * 2;
    if (oi + 1 < out_size) {
      out[oi]     = xsv[tid * 32 + sm];          // Re
      out[oi + 1] = xsv[(tid + 32) * 32 + sm];   // Im
    }
  }
}

extern "C" void kernel_launch(void* const* d_in, const int* in_sizes, int n_in,
                              void* d_out, int out_size, void* d_ws, size_t ws_size,
                              hipStream_t stream) {
  (void)in_sizes; (void)n_in; (void)d_ws; (void)ws_size;
  const double* x  = (const double*)d_in[0];
  const double* y  = (const double*)d_in[1];
  const double* A  = (const double*)d_in[2];
  const double* nv = (const double*)d_in[3];
  float* out = (float*)d_out;
  // zero the atomically-accumulated loss/mse slots (harness poisons d_out)
  mhcg_zero<<<1, 256, 0, stream>>>(out, out_size);
  // one workgroup (8 wave32s) per batch element; all state LDS-resident
  mhcg_kernel<<<BSZ, 256, SMEM_FLOATS * sizeof(float), stream>>>(
      x, y, A, nv, out, out_size);
}